// SphericalGMMLayer_61598420959739
// MI455X (gfx1250) — compile-verified
//
#include <hip/hip_runtime.h>
#include <hip/hip_bf16.h>
#include <math.h>

typedef __attribute__((ext_vector_type(2))) float v2f;
typedef __attribute__((ext_vector_type(8))) float v8f;

#define EPS       1e-5f
#define EPS_SIGMA 1e-3f
#define NB 8
#define NK 32
#define ND 64
#define NP 4096   /* 64*64 pixels */

// ---------------------------------------------------------------------------
// Kernel A: s2[b,p] = sum_d img[b,d,p]^2   (coalesced over p)
// ---------------------------------------------------------------------------
__global__ void k_pixnorm(const float* __restrict__ img, float* __restrict__ s2) {
    int gid = blockIdx.x * 256 + threadIdx.x;           // 0 .. NB*NP-1
    int b = gid >> 12, p = gid & (NP - 1);
    const float* base = img + ((size_t)b * ND) * NP + p;
    float s = 0.f;
#pragma unroll 8
    for (int d = 0; d < ND; ++d) {
        float v = base[(size_t)d * NP];
        s += v * v;
    }
    s2[gid] = s;
}

// ---------------------------------------------------------------------------
// Kernel B: mu_num[b,d,k] = sum_p img[b,d,p] * x[b,k,p]
// GEMM per batch: M=64(d) x N=32(k), K=4096(p).  V_WMMA_F32_16X16X4_F32.
// One block per batch, 8 waves = 4 m-tiles x 2 n-tiles.
// A frag: lane lm row M, half h -> contiguous float2 at col kp+2h. Same for B.
// ---------------------------------------------------------------------------
__global__ void k_gemm_mu(const float* __restrict__ img, const float* __restrict__ x,
                          float* __restrict__ mu_num) {
    int b = blockIdx.x;
    int wave = threadIdx.x >> 5, lane = threadIdx.x & 31;
    int lm = lane & 15, h = lane >> 4;
    int m0 = (wave >> 1) * 16;           // d-tile
    int n0 = (wave & 1) * 16;            // k-tile
    const float* pA = img + ((size_t)(b * ND + m0 + lm)) * NP + 2 * h;
    const float* pB = x   + ((size_t)(b * NK + n0 + lm)) * NP + 2 * h;
    v8f acc = {};
    for (int kp = 0; kp < NP; kp += 16) {
        __builtin_prefetch(pA + kp + 128, 0, 0);   // global_prefetch_b8
        __builtin_prefetch(pB + kp + 128, 0, 0);
#pragma unroll
        for (int j = 0; j < 4; ++j) {
            v2f a  = *(const v2f*)(pA + kp + 4 * j);
            v2f bb = *(const v2f*)(pB + kp + 4 * j);
            acc = __builtin_amdgcn_wmma_f32_16x16x4_f32(
                false, a, false, bb, (short)0, acc, false, false);
        }
    }
#pragma unroll
    for (int r = 0; r < 8; ++r) {
        int d = m0 + r + 8 * h;                        // C/D layout: M = r + 8h
        mu_num[((size_t)(b * ND + d)) * NK + n0 + lm] = acc[r];
    }
}

// ---------------------------------------------------------------------------
// Kernel C: sx[b,k] = sum_p x ;  T1[b,k] = sum_p x * s2   (one wave per (b,k))
// ---------------------------------------------------------------------------
__global__ void k_rowsums(const float* __restrict__ x, const float* __restrict__ s2,
                          float* __restrict__ sx, float* __restrict__ t1) {
    int wid  = (blockIdx.x * blockDim.x + threadIdx.x) >> 5;  // 0..255 = b*32+k
    int lane = threadIdx.x & 31;
    int b = wid >> 5;
    const float* xr  = x  + (size_t)wid * NP;
    const float* s2r = s2 + (size_t)b * NP;
    float s = 0.f, t = 0.f;
    for (int p = lane; p < NP; p += 32) {
        float xv = xr[p];
        s += xv;
        t += xv * s2r[p];
    }
#pragma unroll
    for (int m = 16; m; m >>= 1) {
        s += __shfl_xor(s, m, 32);
        t += __shfl_xor(t, m, 32);
    }
    if (lane == 0) { sx[wid] = s; t1[wid] = t; }
}

// ---------------------------------------------------------------------------
// Kernel D: per-(b,k) stats.
//   mu = mu_num/sxe ; sig_num = T1 - mn2*(2/sxe - sx/sxe^2) ; writes mu_t[b,k,d]
// ---------------------------------------------------------------------------
__global__ void k_stats(const float* __restrict__ mu_num, const float* __restrict__ sx,
                        const float* __restrict__ t1,
                        float* __restrict__ mu_t, float* __restrict__ alpha,
                        float* __restrict__ bcoef, float* __restrict__ mm) {
    int tid = threadIdx.x;               // 256 = b*32+k
    int b = tid >> 5, k = tid & 31;
    float sxv = sx[tid], t1v = t1[tid];
    float sxe = sxv + EPS;
    float inv = 1.0f / sxe;
    float mn2 = 0.f;
    for (int d = 0; d < ND; ++d) {
        float mn = mu_num[((size_t)(b * ND + d)) * NK + k];
        mn2 += mn * mn;
        mu_t[(size_t)tid * ND + d] = mn * inv;          // mu_t[b][k][d]
    }
    float sig_num = t1v - mn2 * (2.0f * inv - sxv * inv * inv);
    float sig = fmaxf(sig_num, 0.0f) / (EPS + sxv * (float)ND);
    alpha[tid] = sxv * (1.0f / (float)NP);
    bcoef[tid] = 0.5f / (sig + EPS_SIGMA);
    mm[tid]    = mn2 * inv * inv;                       // ||mu_k||^2
}

// ---------------------------------------------------------------------------
// Kernel E: fused classify + normalize.
// Per block: one batch, 64-pixel strip. 8 waves = 2 k-tiles x 4 p-tiles.
// dot[k,p] = sum_d mu_t[k,d]*img[d,p] via WMMA (K=64 -> 16 wmma ops/wave).
// val = alpha_k * exp(-bcoef_k * (s2_p - 2*dot + mm_k)); normalize over k in LDS.
// ---------------------------------------------------------------------------
__global__ void k_classify(const float* __restrict__ img, const float* __restrict__ mu_t,
                           const float* __restrict__ s2,
                           const float* __restrict__ alpha, const float* __restrict__ bcoef,
                           const float* __restrict__ mm, float* __restrict__ y) {
    __shared__ float sval[NK][65];       // padded: stride 65 mod 64 == 1
    __shared__ float sA[NK], sB[NK], sM[NK], ss2[64], sinv[64];
    int b = blockIdx.x >> 6, strip = blockIdx.x & 63;
    int pbase = strip * 64;
    int tid = threadIdx.x;
    if (tid < NK) {
        sA[tid] = alpha[b * NK + tid];
        sB[tid] = bcoef[b * NK + tid];
        sM[tid] = mm[b * NK + tid];
    }
    if (tid < 64) ss2[tid] = s2[(size_t)b * NP + pbase + tid];
    __syncthreads();

    int wave = tid >> 5, lane = tid & 31, lm = lane & 15, h = lane >> 4;
    int k0 = (wave >> 2) * 16;           // k-tile base
    int pt = (wave & 3) * 16;            // strip-local pixel tile base
    const float* pA = mu_t + ((size_t)(b * NK + k0 + lm)) * ND + 2 * h;
    const float* pB = img  + ((size_t)(b * ND)) * NP + pbase + pt + lm;
    v8f acc = {};
#pragma unroll
    for (int d0 = 0; d0 < ND; d0 += 4) {
        v2f a = *(const v2f*)(pA + d0);          // mu_t contiguous in d
        v2f bb;
        bb.x = pB[(size_t)(d0 + 2 * h) * NP];    // img column, coalesced over lanes
        bb.y = pB[(size_t)(d0 + 2 * h + 1) * NP];
        acc = __builtin_amdgcn_wmma_f32_16x16x4_f32(
            false, a, false, bb, (short)0, acc, false, false);
    }
#pragma unroll
    for (int r = 0; r < 8; ++r) {
        int k  = k0 + r + 8 * h;
        int pl = pt + lm;
        float iz2 = ss2[pl] - 2.0f * acc[r] + sM[k];
        sval[k][pl] = sA[k] * __expf(-sB[k] * iz2);
    }
    __syncthreads();
    if (tid < 64) {
        float s = 0.f;
#pragma unroll
        for (int k = 0; k < NK; ++k) s += sval[k][tid];
        sinv[tid] = 1.0f / (EPS + s);
    }
    __syncthreads();
#pragma unroll
    for (int i = 0; i < 8; ++i) {
        int e = i * 256 + tid;
        int k = e >> 6, p = e & 63;
        y[((size_t)(b * NK + k)) * NP + pbase + p] = sval[k][p] * sinv[p];
    }
}

// ---------------------------------------------------------------------------
extern "C" void kernel_launch(void* const* d_in, const int* in_sizes, int n_in,
                              void* d_out, int out_size, void* d_ws, size_t ws_size,
                              hipStream_t stream) {
    const float* x   = (const float*)d_in[0];   // [8,32,64,64]
    const float* img = (const float*)d_in[1];   // [8,64,64,64]
    float* y = (float*)d_out;                   // [8,32,64,64]

    float* ws = (float*)d_ws;
    float* s2     = ws;                         // 8*4096   = 32768
    float* mu_num = s2     + NB * NP;           // 8*64*32  = 16384
    float* mu_t   = mu_num + NB * ND * NK;      // 8*32*64  = 16384
    float* sx     = mu_t   + NB * NK * ND;      // 256
    float* t1     = sx     + NB * NK;           // 256
    float* alpha  = t1     + NB * NK;           // 256
    float* bcoef  = alpha  + NB * NK;           // 256
    float* mm     = bcoef  + NB * NK;           // 256

    k_pixnorm <<<(NB * NP) / 256, 256, 0, stream>>>(img, s2);
    k_gemm_mu <<<NB, 256, 0, stream>>>(img, x, mu_num);
    k_rowsums <<<(NB * NK * 32) / 256, 256, 0, stream>>>(x, s2, sx, t1);
    k_stats   <<<1, NB * NK, 0, stream>>>(mu_num, sx, t1, mu_t, alpha, bcoef, mm);
    k_classify<<<NB * (NP / 64), 256, 0, stream>>>(img, mu_t, s2, alpha, bcoef, mm, y);
}